// GraphMemoryQuery_36051955483060
// MI455X (gfx1250) — compile-verified
//
#include <hip/hip_runtime.h>
#include <math.h>

// Shapes from the reference
#define BB 64
#define SS 1024
#define VV 1024
#define DD 1024

#define NCH 16               // s-chunks per batch for the GEMV split
#define ROWS (SS / NCH)      // 64 rows per chunk

typedef __attribute__((ext_vector_type(2))) float v2f;
typedef __attribute__((ext_vector_type(8))) float v8f;

__device__ __forceinline__ float gelu_tanh(float x) {
  // jax.nn.gelu default (approximate=True)
  float x3 = x * x * x;
  float t = tanhf(0.7978845608028654f * (x + 0.044715f * x3));
  return 0.5f * x * (1.0f + t);
}

// ---------------------------------------------------------------------------
// K1: w1[b,:] = mask[b] * step[b, qi[b], :];  wsum = e_q*mask + w1
// ---------------------------------------------------------------------------
__global__ void k_gather(const float* __restrict__ step,
                         const int* __restrict__ qidx,
                         const float* __restrict__ qmask,
                         float* __restrict__ w_cur,
                         float* __restrict__ wsum) {
  int b = blockIdx.x;
  int qi = qidx[b];
  float m = qmask[b];
  const float* row = step + (size_t)b * SS * SS + (size_t)qi * SS;
  for (int t = threadIdx.x; t < SS; t += blockDim.x) {
    float w1 = m * row[t];
    w_cur[b * SS + t] = w1;
    wsum[b * SS + t] = w1 + ((t == qi) ? m : 0.0f);
  }
}

// ---------------------------------------------------------------------------
// Per-batch GEMV stage 1: partial[b,ch,:] = sum_{s in chunk ch} w[b,s]*M[b,s,:]
// Grid: B*NCH blocks x 256 threads. Each thread owns 4 consecutive columns
// (float4 / b128 loads); a block covers the full 4KB row, streaming 64 rows.
// 1024 blocks (8192 waves), unroll 8 -> deep MLP for the 23.3 TB/s HBM.
// ---------------------------------------------------------------------------
__global__ void k_gemv_part(const float* __restrict__ w_in,
                            const float* __restrict__ M,
                            float* __restrict__ partial) {
  __shared__ float lw[ROWS];
  int b = blockIdx.x >> 4;
  int ch = blockIdx.x & (NCH - 1);
  int tid = threadIdx.x;
  if (tid < ROWS) lw[tid] = w_in[b * SS + ch * ROWS + tid];
  __syncthreads();

  const float4* Mrow =
      (const float4*)(M + (size_t)b * SS * VV + (size_t)(ch * ROWS) * VV) + tid;
  float4 acc = make_float4(0.f, 0.f, 0.f, 0.f);
#pragma unroll 8
  for (int s = 0; s < ROWS; ++s) {
    float4 v = Mrow[(size_t)s * (VV / 4)];
    float w = lw[s];
    acc.x = fmaf(w, v.x, acc.x);
    acc.y = fmaf(w, v.y, acc.y);
    acc.z = fmaf(w, v.z, acc.z);
    acc.w = fmaf(w, v.w, acc.w);
  }
  ((float4*)partial)[(size_t)(b * NCH + ch) * (VV / 4) + tid] = acc;
}

// ---------------------------------------------------------------------------
// GEMV stage 2: out[b,:] = sum_ch partial[b,ch,:]  (fixed order: deterministic)
// Optionally wsum[b,:] += out[b,:]  (unique owner per element, no races).
// ---------------------------------------------------------------------------
template <bool ACCUM>
__global__ void k_gemv_reduce(const float* __restrict__ partial,
                              float* __restrict__ out,
                              float* __restrict__ accum) {
  int b = blockIdx.x;
  int tid = threadIdx.x;
  const float4* p = (const float4*)partial + (size_t)b * NCH * (VV / 4) + tid;
  float4 s = make_float4(0.f, 0.f, 0.f, 0.f);
#pragma unroll
  for (int ch = 0; ch < NCH; ++ch) {
    float4 v = p[(size_t)ch * (VV / 4)];
    s.x += v.x;
    s.y += v.y;
    s.z += v.z;
    s.w += v.w;
  }
  ((float4*)out)[(size_t)b * (VV / 4) + tid] = s;
  if (ACCUM) {
    float4* ap = (float4*)accum + (size_t)b * (VV / 4) + tid;
    float4 a = *ap;
    a.x += s.x;
    a.y += s.y;
    a.z += s.z;
    a.w += s.w;
    *ap = a;
  }
}

// ---------------------------------------------------------------------------
// WMMA fragment layout (V_WMMA_F32_16X16X4_F32):
//   A 16x4 : lane&15 = M row; (lane>>4)*2 = K offset; v2f = (K, K+1)
//   B 4x16 : lane&15 = N col; (lane>>4)*2 = K offset; v2f = (K, K+1)
//   C 16x16: VGPR r, lanes 0-15 -> M=r, lanes 16-31 -> M=r+8; N = lane&15
// ---------------------------------------------------------------------------

// K4: gs = wsum @ symbol_bank + accv @ value_bank   ([64,1024]x[1024,1024] x2)
__global__ void k_gs(const float* __restrict__ wsum,
                     const float* __restrict__ sym,
                     const float* __restrict__ accv,
                     const float* __restrict__ val,
                     float* __restrict__ gs) {
  int lane = threadIdx.x & 31;
  int wid = (blockIdx.x << 3) | (threadIdx.x >> 5);  // 32 blocks * 8 waves
  int m0 = (wid >> 6) << 4;                          // 4 M-tiles
  int n0 = (wid & 63) << 4;                          // 64 N-tiles
  int arow = lane & 15;
  int koff = (lane >> 4) << 1;
  int col = lane & 15;

  const float* ap1 = wsum + (size_t)(m0 + arow) * SS + koff;
  const float* ap2 = accv + (size_t)(m0 + arow) * VV + koff;
  const float* bp1 = sym + (size_t)koff * DD + n0 + col;
  const float* bp2 = val + (size_t)koff * DD + n0 + col;

  v8f c = {};
#pragma unroll 4
  for (int k = 0; k < 1024; k += 4) {
    v2f a, b;
    a.x = ap1[k];
    a.y = ap1[k + 1];
    b.x = bp1[(size_t)k * DD];
    b.y = bp1[(size_t)k * DD + DD];
    c = __builtin_amdgcn_wmma_f32_16x16x4_f32(false, a, false, b, (short)0, c,
                                              false, false);
    a.x = ap2[k];
    a.y = ap2[k + 1];
    b.x = bp2[(size_t)k * DD];
    b.y = bp2[(size_t)k * DD + DD];
    c = __builtin_amdgcn_wmma_f32_16x16x4_f32(false, a, false, b, (short)0, c,
                                              false, false);
  }
  int crow = (lane >> 4) << 3;
#pragma unroll
  for (int r = 0; r < 8; ++r)
    gs[(size_t)(m0 + crow + r) * DD + n0 + col] = c[r];
}

// K5/K6: C = (gelu?)(A @ W + bias)   A:[64,1024] W:[1024,1024] C:[64,1024]
template <bool GELU>
__global__ void k_gemm(const float* __restrict__ A,
                       const float* __restrict__ W,
                       const float* __restrict__ bias,
                       float* __restrict__ C) {
  int lane = threadIdx.x & 31;
  int wid = (blockIdx.x << 3) | (threadIdx.x >> 5);
  int m0 = (wid >> 6) << 4;
  int n0 = (wid & 63) << 4;
  int arow = lane & 15;
  int koff = (lane >> 4) << 1;
  int col = lane & 15;

  const float* ap = A + (size_t)(m0 + arow) * DD + koff;
  const float* bp = W + (size_t)koff * DD + n0 + col;

  v8f c = {};
#pragma unroll 4
  for (int k = 0; k < 1024; k += 4) {
    v2f a, b;
    a.x = ap[k];
    a.y = ap[k + 1];
    b.x = bp[(size_t)k * DD];
    b.y = bp[(size_t)k * DD + DD];
    c = __builtin_amdgcn_wmma_f32_16x16x4_f32(false, a, false, b, (short)0, c,
                                              false, false);
  }
  float bv = bias[n0 + col];
  int crow = (lane >> 4) << 3;
#pragma unroll
  for (int r = 0; r < 8; ++r) {
    float v = c[r] + bv;
    if (GELU) v = gelu_tanh(v);
    C[(size_t)(m0 + crow + r) * DD + n0 + col] = v;
  }
}

// ---------------------------------------------------------------------------
extern "C" void kernel_launch(void* const* d_in, const int* in_sizes, int n_in,
                              void* d_out, int out_size, void* d_ws,
                              size_t ws_size, hipStream_t stream) {
  const float* map_memory = (const float*)d_in[0];   // [B,S,V]
  const float* step_memory = (const float*)d_in[1];  // [B,S,S]
  const int* query_idx = (const int*)d_in[2];        // [B]
  const float* query_mask = (const float*)d_in[3];   // [B]
  const float* symbol_bank = (const float*)d_in[4];  // [S,D]
  const float* value_bank = (const float*)d_in[5];   // [V,D]
  const float* Wo1 = (const float*)d_in[6];
  const float* bo1 = (const float*)d_in[7];
  const float* Wo2 = (const float*)d_in[8];
  const float* bo2 = (const float*)d_in[9];
  const float* Wf1 = (const float*)d_in[10];
  const float* bf1 = (const float*)d_in[11];
  const float* Wf2 = (const float*)d_in[12];
  const float* bf2 = (const float*)d_in[13];
  float* out = (float*)d_out;  // [B,V] logits ++ [B,D] feedback

  float* ws = (float*)d_ws;
  const int T = BB * SS;  // 65536 floats per [B,*] buffer
  float* w_cur = ws;
  float* w_next = ws + T;
  float* wsum = ws + 2 * T;
  float* accv = ws + 3 * T;
  float* gs = ws + 4 * T;
  float* h1 = ws + 5 * T;
  float* h2 = ws + 6 * T;
  float* part = ws + 7 * T;  // [B, NCH, V] = 4 MB of partials

  // Walk (sequential dependency): w1 gather, then two dense vec-mat passes
  k_gather<<<BB, 256, 0, stream>>>(step_memory, query_idx, query_mask, w_cur,
                                   wsum);
  k_gemv_part<<<BB * NCH, 256, 0, stream>>>(w_cur, step_memory, part);
  k_gemv_reduce<true><<<BB, 256, 0, stream>>>(part, w_next, wsum);
  k_gemv_part<<<BB * NCH, 256, 0, stream>>>(w_next, step_memory, part);
  k_gemv_reduce<true><<<BB, 256, 0, stream>>>(part, w_cur, wsum);

  // acc_values = wsum @ map_memory[b]  (one streaming pass over map_memory)
  k_gemv_part<<<BB * NCH, 256, 0, stream>>>(wsum, map_memory, part);
  k_gemv_reduce<false><<<BB, 256, 0, stream>>>(part, accv, nullptr);

  // graph_state = wsum @ symbol_bank + acc_values @ value_bank
  k_gs<<<32, 256, 0, stream>>>(wsum, symbol_bank, accv, value_bank, gs);

  // heads
  k_gemm<true><<<32, 256, 0, stream>>>(gs, Wo1, bo1, h1);
  k_gemm<true><<<32, 256, 0, stream>>>(gs, Wf1, bf1, h2);
  k_gemm<false><<<32, 256, 0, stream>>>(h1, Wo2, bo2, out);            // logits
  k_gemm<false><<<32, 256, 0, stream>>>(h2, Wf2, bf2, out + BB * DD);  // feedback
}